// SelfAtt_17197049053534
// MI455X (gfx1250) — compile-verified
//
#include <hip/hip_runtime.h>
#include <math.h>

#define B     8
#define C     512
#define HW    4096      // h*w = 64*64
#define HEADS 8
#define DH    64        // head dim
#define OC3   1536      // 3*C
#define G     32        // norm groups
#define CPG   16        // channels per group
#define GN_EPS 1e-5f

typedef float v2f __attribute__((ext_vector_type(2)));
typedef float v8f __attribute__((ext_vector_type(8)));

// fp32 WMMA: D(16x16) = A(16x4) * B(4x16) + C   (exact fp32 — workload is
// memory-bound on MI455X, so we keep full precision instead of bf16/fp8)
__device__ __forceinline__ v8f wmma4(v2f a, v2f b, v8f c) {
  return __builtin_amdgcn_wmma_f32_16x16x4_f32(
      /*neg_a=*/false, a, /*neg_b=*/false, b,
      /*c_mod=*/(short)0, c, /*reuse_a=*/false, /*reuse_b=*/false);
}

// ---------------------------------------------------------------------------
// Kernel 1: GroupNorm. One block per (b, group). 16 ch x 4096 = 65536 elems.
// ---------------------------------------------------------------------------
__global__ void gn_kernel(const float* __restrict__ x, const float* __restrict__ gnw,
                          const float* __restrict__ gnb, float* __restrict__ xn) {
  const int bg = blockIdx.x;            // 0..255
  const int b = bg >> 5, g = bg & 31;
  const long base = (long)b * C * HW + (long)g * CPG * HW;
  const int tid = threadIdx.x;
  const int NE = CPG * HW;              // 65536
  float s = 0.f, ss = 0.f;
  for (int i = tid; i < NE; i += 256) {
    float v = x[base + i];
    s += v; ss += v * v;
  }
  __shared__ float sh_s[256], sh_q[256];
  sh_s[tid] = s; sh_q[tid] = ss;
  __syncthreads();
  for (int off = 128; off > 0; off >>= 1) {
    if (tid < off) { sh_s[tid] += sh_s[tid + off]; sh_q[tid] += sh_q[tid + off]; }
    __syncthreads();
  }
  const float mean = sh_s[0] * (1.0f / NE);
  const float var  = sh_q[0] * (1.0f / NE) - mean * mean;
  const float inv  = rsqrtf(var + GN_EPS);
  for (int i = tid; i < NE; i += 256) {
    const int ch = g * CPG + (i >> 12);                // i / 4096
    const float v = (x[base + i] - mean) * inv;
    xn[base + i] = v * gnw[ch] + gnb[ch];
  }
}

// ---------------------------------------------------------------------------
// Kernel 2: QKV GEMM.  Y[b][o][n] = sum_c W[o][c] * X[b][c][n]
// grid (HW/64, OC3/128, B), block 256 (8 waves). Wave tile 16(M) x 64(N).
// ---------------------------------------------------------------------------
__global__ void gemm_qkv(const float* __restrict__ W, const float* __restrict__ X,
                         float* __restrict__ Y) {
  const int lane  = threadIdx.x & 31;
  const int wave  = threadIdx.x >> 5;
  const int mlane = lane & 15;          // M row (A) / N col (B,C)
  const int kp    = lane >> 4;          // K-pair selector
  const int n0 = blockIdx.x * 64;
  const int m0 = blockIdx.y * 128 + wave * 16;
  const int b  = blockIdx.z;
  const float* Xb = X + (long)b * C * HW;
  v8f acc[4] = {};
  for (int k0 = 0; k0 < C; k0 += 4) {
    v2f a;
    const float* wp = W + (m0 + mlane) * C + k0 + 2 * kp;
    a.x = wp[0]; a.y = wp[1];
    const float* xr = Xb + (long)(k0 + 2 * kp) * HW;
#pragma unroll
    for (int j = 0; j < 4; ++j) {
      v2f bb;
      bb.x = xr[n0 + 16 * j + mlane];
      bb.y = xr[HW + n0 + 16 * j + mlane];
      acc[j] = wmma4(a, bb, acc[j]);
    }
  }
  float* Yb = Y + (long)b * OC3 * HW;
#pragma unroll
  for (int j = 0; j < 4; ++j)
#pragma unroll
    for (int r = 0; r < 8; ++r)
      Yb[(long)(m0 + r + 8 * kp) * HW + n0 + 16 * j + mlane] = acc[j][r];
}

// ---------------------------------------------------------------------------
// Kernel 3: softmax over n=4096 for each row of the K slice (channels C..2C).
// One block per row (B*C = 4096 rows).
// ---------------------------------------------------------------------------
__global__ void softmax_k(float* __restrict__ qkv) {
  const int row = blockIdx.x;                // 0..4095
  const int b = row >> 9, oc = row & 511;
  float* p = qkv + (long)b * OC3 * HW + (long)(C + oc) * HW;
  const int tid = threadIdx.x;
  __shared__ float sh[256];
  float m = -3.4e38f;
  for (int i = tid; i < HW; i += 256) m = fmaxf(m, p[i]);
  sh[tid] = m; __syncthreads();
  for (int off = 128; off > 0; off >>= 1) {
    if (tid < off) sh[tid] = fmaxf(sh[tid], sh[tid + off]);
    __syncthreads();
  }
  m = sh[0]; __syncthreads();
  float s = 0.f;
  for (int i = tid; i < HW; i += 256) {
    float e = __expf(p[i] - m);
    p[i] = e; s += e;
  }
  sh[tid] = s; __syncthreads();
  for (int off = 128; off > 0; off >>= 1) {
    if (tid < off) sh[tid] += sh[tid + off];
    __syncthreads();
  }
  const float inv = 1.f / sh[0];
  for (int i = tid; i < HW; i += 256) p[i] *= inv;
}

// ---------------------------------------------------------------------------
// Kernel 4: att[bh][d][e] = sum_n K[d][n] * V[e][n].  K-dim = n = 4096.
// grid B*HEADS, block 256 (8 waves, 2 of 16 tiles each).
// ---------------------------------------------------------------------------
__global__ void att_kernel(const float* __restrict__ qkv, float* __restrict__ att) {
  const int bh = blockIdx.x;
  const int b = bh >> 3, h = bh & 7;
  const float* Kt = qkv + (long)b * OC3 * HW + (long)(C + h * DH) * HW;
  const float* Vt = qkv + (long)b * OC3 * HW + (long)(2 * C + h * DH) * HW;
  const int lane = threadIdx.x & 31;
  const int wave = threadIdx.x >> 5;
  const int mlane = lane & 15, kp = lane >> 4;
  const int d0 = (wave >> 1) * 16;
  const int e0 = (wave & 1) * 32;
  v8f acc[2] = {};
  for (int n = 0; n < HW; n += 4) {
    v2f a;
    const float* kr = Kt + (long)(d0 + mlane) * HW + n + 2 * kp;
    a.x = kr[0]; a.y = kr[1];
#pragma unroll
    for (int jj = 0; jj < 2; ++jj) {
      const float* vr = Vt + (long)(e0 + 16 * jj + mlane) * HW + n + 2 * kp;
      v2f bb; bb.x = vr[0]; bb.y = vr[1];
      acc[jj] = wmma4(a, bb, acc[jj]);
    }
  }
  float* ap = att + (long)bh * DH * DH;
#pragma unroll
  for (int jj = 0; jj < 2; ++jj)
#pragma unroll
    for (int r = 0; r < 8; ++r)
      ap[(d0 + r + 8 * kp) * DH + e0 + 16 * jj + mlane] = acc[jj][r];
}

// ---------------------------------------------------------------------------
// Kernel 5: out[b][h*64+e][n] = sum_d att[d][e] * Q[d][n].  K-dim = d = 64.
// grid (HW/256, B*HEADS), block 256. Wave tile 16(M) x 128(N), 8 accums.
// ---------------------------------------------------------------------------
__global__ void out_kernel(const float* __restrict__ qkv, const float* __restrict__ att,
                           float* __restrict__ outb) {
  const int bh = blockIdx.y;
  const int b = bh >> 3, h = bh & 7;
  const float* Qt = qkv + (long)b * OC3 * HW + (long)(h * DH) * HW;
  const float* ap = att + (long)bh * DH * DH;
  const int lane = threadIdx.x & 31;
  const int wave = threadIdx.x >> 5;
  const int mlane = lane & 15, kp = lane >> 4;
  const int m0 = (wave & 3) * 16;
  const int nbase = blockIdx.x * 256 + (wave >> 2) * 128;
  v8f acc[8] = {};
  for (int k0 = 0; k0 < DH; k0 += 4) {
    v2f a;   // A[m=e][k=d] = att[d][e] (transposed read)
    a.x = ap[(k0 + 2 * kp) * DH + m0 + mlane];
    a.y = ap[(k0 + 2 * kp + 1) * DH + m0 + mlane];
    const float* qr = Qt + (long)(k0 + 2 * kp) * HW;
#pragma unroll
    for (int j = 0; j < 8; ++j) {
      v2f bb;
      bb.x = qr[nbase + 16 * j + mlane];
      bb.y = qr[HW + nbase + 16 * j + mlane];
      acc[j] = wmma4(a, bb, acc[j]);
    }
  }
  float* op = outb + (long)b * C * HW + (long)(h * DH) * HW;
#pragma unroll
  for (int j = 0; j < 8; ++j)
#pragma unroll
    for (int r = 0; r < 8; ++r)
      op[(long)(m0 + r + 8 * kp) * HW + nbase + 16 * j + mlane] = acc[j][r];
}

// ---------------------------------------------------------------------------
// Kernel 6: y = x + Wp @ out + bias.  grid (HW/64, C/128, B), block 256.
// ---------------------------------------------------------------------------
__global__ void proj_kernel(const float* __restrict__ Wp, const float* __restrict__ X,
                            const float* __restrict__ outb, const float* __restrict__ bias,
                            float* __restrict__ Y) {
  const int lane = threadIdx.x & 31;
  const int wave = threadIdx.x >> 5;
  const int mlane = lane & 15, kp = lane >> 4;
  const int n0 = blockIdx.x * 64;
  const int m0 = blockIdx.y * 128 + wave * 16;
  const int b  = blockIdx.z;
  const float* Ob = outb + (long)b * C * HW;
  v8f acc[4] = {};
  for (int k0 = 0; k0 < C; k0 += 4) {
    v2f a;
    const float* wp = Wp + (m0 + mlane) * C + k0 + 2 * kp;
    a.x = wp[0]; a.y = wp[1];
    const float* xr = Ob + (long)(k0 + 2 * kp) * HW;
#pragma unroll
    for (int j = 0; j < 4; ++j) {
      v2f bb;
      bb.x = xr[n0 + 16 * j + mlane];
      bb.y = xr[HW + n0 + 16 * j + mlane];
      acc[j] = wmma4(a, bb, acc[j]);
    }
  }
  const float* Xb = X + (long)b * C * HW;
  float* Yb = Y + (long)b * C * HW;
#pragma unroll
  for (int j = 0; j < 4; ++j)
#pragma unroll
    for (int r = 0; r < 8; ++r) {
      const int m = m0 + r + 8 * kp;
      const long idx = (long)m * HW + n0 + 16 * j + mlane;
      Yb[idx] = Xb[idx] + acc[j][r] + bias[m];
    }
}

// ---------------------------------------------------------------------------
extern "C" void kernel_launch(void* const* d_in, const int* in_sizes, int n_in,
                              void* d_out, int out_size, void* d_ws, size_t ws_size,
                              hipStream_t stream) {
  (void)in_sizes; (void)n_in; (void)out_size; (void)ws_size;
  const float* x     = (const float*)d_in[0];
  const float* gnw   = (const float*)d_in[1];
  const float* gnb   = (const float*)d_in[2];
  const float* qkvw  = (const float*)d_in[3];
  const float* projw = (const float*)d_in[4];
  const float* projb = (const float*)d_in[5];
  float* y = (float*)d_out;

  char* ws = (char*)d_ws;
  const size_t XN_BYTES  = (size_t)B * C * HW * 4;     // 64 MiB (reused as "out")
  const size_t QKV_BYTES = (size_t)B * OC3 * HW * 4;   // 192 MiB
  float* xn  = (float*)(ws);
  float* qkv = (float*)(ws + XN_BYTES);
  float* att = (float*)(ws + XN_BYTES + QKV_BYTES);    // 1 MiB

  gn_kernel  <<<B * G, 256, 0, stream>>>(x, gnw, gnb, xn);
  gemm_qkv   <<<dim3(HW / 64, OC3 / 128, B), 256, 0, stream>>>(qkvw, xn, qkv);
  softmax_k  <<<B * C, 256, 0, stream>>>(qkv);
  att_kernel <<<B * HEADS, 256, 0, stream>>>(qkv, att);
  out_kernel <<<dim3(HW / 256, B * HEADS), 256, 0, stream>>>(qkv, att, xn); // xn -> out
  proj_kernel<<<dim3(HW / 64, C / 128, B), 256, 0, stream>>>(projw, x, xn, projb, y);
}